// MambaMixer_69380901700057
// MI455X (gfx1250) — compile-verified
//
#include <hip/hip_runtime.h>
#include <hip/hip_bf16.h>
#include <cstddef>

// ---------------------------------------------------------------------------
// Mamba mixer on gfx1250 (CDNA5).  B=1, S=2048, H=1024, INTER=2048,
// K=4 (conv), DST=16, DTR=64.  fp32 throughout, GEMMs on V_WMMA_F32_16X16X4_F32
// with async global->LDS double-buffered B staging (GLOBAL_LOAD_ASYNC_TO_LDS)
// and hand-paired DS_LOAD_2ADDR_B32 fragment reads.
// ---------------------------------------------------------------------------

#define S_LEN   2048
#define HID     1024
#define INTER   2048
#define KCONV   4
#define DST     16
#define DTR     64
#define NPAD    128          // W_x columns (96) padded to 128
#define KTILE   32           // K-depth staged per LDS tile (8 WMMA steps)

typedef float v2f __attribute__((ext_vector_type(2)));
typedef float v8f __attribute__((ext_vector_type(8)));

// ---------------------------------------------------------------------------
// Generic fp32 GEMM: C[M,N] = A[M,K] * B[K,N], all row-major.
// Block = 256 threads (8 wave32s).  Each wave computes a 16x64 tile via four
// 16x16 accumulators; K consumed 4 at a time with V_WMMA_F32_16X16X4_F32.
// Requires M%128==0, N%64==0, K%KTILE==0 (all call sites satisfy this).
//
// B tiles (KTILE x 64 = 8 KB) are shared by all 8 waves, so they are staged
// once per block through LDS using async copies (ASYNCcnt-tracked), double
// buffered:  issue(t+1) -> s_wait_asynccnt 2 -> barrier -> WMMA from LDS.
//
// A-fragment (16x4 f32): lanes 0-15 hold row m=lane, k={k0,k0+1};
//                        lanes 16-31 hold row m=lane-16, k={k0+2,k0+3}.
// B-fragment (4x16 f32): lanes 0-15 hold col n=lane, rows {k0,k0+1};
//                        lanes 16-31 hold col n=lane-16, rows {k0+2,k0+3}.
//   -> per lane the pair {B[r][n], B[r+1][n]} sits 256 B apart in the LDS
//      tile: fetched with one ds_load_2addr_b32 into a contiguous VGPR pair.
// C/D (16x16 f32, 8 VGPRs): vgpr i -> row i (lanes 0-15) / row 8+i (16-31).
// ---------------------------------------------------------------------------
__global__ __launch_bounds__(256)
void gemm_wmma_f32(const float* __restrict__ A, int lda,
                   const float* __restrict__ B, int ldb,
                   float* __restrict__ C, int ldc,
                   int K) {
  __shared__ float btile[2][KTILE * 64];      // 2 x 8 KB double buffer

  const int tid  = threadIdx.x;
  const int lane = tid & 31;
  const int wave = tid >> 5;
  const int half = lane >> 4;          // 0 or 1
  const int l16  = lane & 15;

  const int m0 = blockIdx.y * 128 + wave * 16;
  const int n0 = blockIdx.x * 64;

  // cooperative B loader: thread -> (row = tid/16, 4 consecutive cols); two
  // b128 async copies per thread cover the full KTILE x 64 tile.
  const int brow = tid >> 4;                       // 0..15
  const int bcol = (tid & 15) << 2;                // 0,4,...,60
  const float* Bg = B + (size_t)brow * ldb + n0 + bcol;
  const size_t bstep16 = (size_t)16 * ldb;         // rows 16..31 of the tile

  // LDS byte offsets (flat LDS aperture keeps offset in addr[31:0])
  const unsigned ldsw0 = (unsigned)(size_t)(&btile[0][brow * 64 + bcol]);
  const unsigned ldsw1 = (unsigned)(size_t)(&btile[1][brow * 64 + bcol]);
  // per-lane LDS read base for fragment fetches (byte address)
  const unsigned ldsr0 = (unsigned)(size_t)(&btile[0][half * 2 * 64 + l16]);
  const unsigned ldsr1 = (unsigned)(size_t)(&btile[1][half * 2 * 64 + l16]);

#define ASYNC_COPY_TILE(tidx, bufsel)                                         \
  do {                                                                        \
    const float* _g = Bg + (size_t)(tidx) * KTILE * ldb;                      \
    const unsigned _l = (bufsel) ? ldsw1 : ldsw0;                             \
    asm volatile("global_load_async_to_lds_b128 %0, %1, off"                  \
                 :: "v"(_l), "v"(_g) : "memory");                             \
    asm volatile("global_load_async_to_lds_b128 %0, %1, off"                  \
                 :: "v"(_l + 4096u), "v"(_g + bstep16) : "memory");           \
  } while (0)

// one WMMA B-fragment: {B[r][n], B[r+1][n]} = LDS pair 64 floats apart,
// directly into a contiguous VGPR pair (offsets in 4-byte units)
#define LOAD_BFRAG(dst, baddr, o0, o1)                                        \
  asm volatile("ds_load_2addr_b32 %0, %1 offset0:" #o0 " offset1:" #o1        \
               : "=v"(dst) : "v"(baddr))

  v8f acc0 = {}, acc1 = {}, acc2 = {}, acc3 = {};

  const float* Arow = A + (size_t)(m0 + l16) * lda + half * 2;
  const int ntiles = K / KTILE;

  ASYNC_COPY_TILE(0, 0);                           // prologue

  for (int t = 0; t < ntiles; ++t) {
    if (t + 1 < ntiles) {
      ASYNC_COPY_TILE(t + 1, (t + 1) & 1);
      asm volatile("s_wait_asynccnt 0x2" ::: "memory");  // tile t complete
    } else {
      asm volatile("s_wait_asynccnt 0x0" ::: "memory");
    }
    __syncthreads();                               // tile t visible to all waves

    const unsigned ldsr = (t & 1) ? ldsr1 : ldsr0;
    const float* Ak = Arow + t * KTILE;
    __builtin_prefetch(Ak + KTILE, 0, 0);          // global_prefetch_b8

#pragma unroll
    for (int kk = 0; kk < KTILE; kk += 4) {
      v2f a = *(const v2f*)(Ak + kk);              // contiguous -> global_load_b64

      const unsigned baddr = ldsr + (unsigned)(kk * 64 * 4);
      v2f b0, b1, b2, b3;
      LOAD_BFRAG(b0, baddr, 0, 64);
      LOAD_BFRAG(b1, baddr, 16, 80);
      LOAD_BFRAG(b2, baddr, 32, 96);
      LOAD_BFRAG(b3, baddr, 48, 112);
      // single DScnt drain; fragments flow through so WMMA can't hoist above
      asm volatile("s_wait_dscnt 0x0"
                   : "+v"(b0), "+v"(b1), "+v"(b2), "+v"(b3) :: "memory");

      acc0 = __builtin_amdgcn_wmma_f32_16x16x4_f32(false, a, false, b0, (short)0, acc0, false, false);
      acc1 = __builtin_amdgcn_wmma_f32_16x16x4_f32(false, a, false, b1, (short)0, acc1, false, false);
      acc2 = __builtin_amdgcn_wmma_f32_16x16x4_f32(false, a, false, b2, (short)0, acc2, false, false);
      acc3 = __builtin_amdgcn_wmma_f32_16x16x4_f32(false, a, false, b3, (short)0, acc3, false, false);
    }
    __syncthreads();                               // protect buf before overwrite
  }
#undef ASYNC_COPY_TILE
#undef LOAD_BFRAG

  float* Crow = C + (size_t)(m0 + half * 8) * ldc + n0 + l16;
#pragma unroll
  for (int i = 0; i < 8; ++i) {
    Crow[(size_t)i * ldc + 0]  = acc0[i];
    Crow[(size_t)i * ldc + 16] = acc1[i];
    Crow[(size_t)i * ldc + 32] = acc2[i];
    Crow[(size_t)i * ldc + 48] = acc3[i];
  }
}

// ---------------------------------------------------------------------------
// Causal depthwise conv (K=4) over time + bias + SiLU.
// raw = proj[:, 0:INTER] (ld 2*INTER).  One thread per (s, c).
// ---------------------------------------------------------------------------
__global__ __launch_bounds__(256)
void conv_silu_kernel(const float* __restrict__ proj,
                      const float* __restrict__ conv_w,
                      const float* __restrict__ conv_b,
                      float* __restrict__ hbuf) {
  const int idx = blockIdx.x * blockDim.x + threadIdx.x;   // [0, S*INTER)
  const int c = idx & (INTER - 1);
  const int s = idx >> 11;                                 // INTER == 2048
  float acc = conv_b[c];
#pragma unroll
  for (int k = 0; k < KCONV; ++k) {
    const int t = s - (KCONV - 1) + k;
    if (t >= 0) acc += proj[(size_t)t * (2 * INTER) + c] * conv_w[k * INTER + c];
  }
  const float sig = 1.0f / (1.0f + __expf(-acc));
  hbuf[idx] = acc * sig;
}

// W_x (INTER x 96) -> zero-padded (INTER x 128)
__global__ __launch_bounds__(256)
void pad_wx_kernel(const float* __restrict__ Wx, float* __restrict__ Wxp) {
  const int idx = blockIdx.x * blockDim.x + threadIdx.x;   // [0, INTER*NPAD)
  const int col = idx & (NPAD - 1);
  const int row = idx >> 7;
  Wxp[idx] = (col < (DTR + 2 * DST)) ? Wx[row * (DTR + 2 * DST) + col] : 0.0f;
}

// dt = softplus(dt_pre + b_dt), in place
__global__ __launch_bounds__(256)
void softplus_bias_kernel(float* __restrict__ dt, const float* __restrict__ b_dt) {
  const int idx = blockIdx.x * blockDim.x + threadIdx.x;   // [0, S*INTER)
  const int c = idx & (INTER - 1);
  const float v = dt[idx] + b_dt[c];
  dt[idx] = (v > 20.0f) ? v : __logf(1.0f + __expf(v));
}

// ---------------------------------------------------------------------------
// Fused selective-scan.  One lane per (channel, d):  2048 ch x 16 d = 1024
// wave32s.  Each wave handles 2 channels (lanes 0-15 / 16-31).  Per step:
//   dA     = exp(dt * A[c,d])            (A = -exp(A_log))
//   state  = dA*state + dt*B[s,d]*u
//   y      = sum_d state*C[s,d]          (shfl_xor tree over 16 lanes)
//   out    = (y + u*D[c]) * silu(gate)   (fused D-skip and gate)
// dA/dBu are never materialized (would be 268 MB each).
// ---------------------------------------------------------------------------
__global__ __launch_bounds__(256)
void ssm_scan_kernel(const float* __restrict__ ssmp,   // (S, 128): [ts | B | C | pad]
                     const float* __restrict__ dt,     // (S, INTER)
                     const float* __restrict__ hbuf,   // (S, INTER)
                     const float* __restrict__ proj,   // (S, 2*INTER) for gate
                     const float* __restrict__ A_log,  // (INTER, DST)
                     const float* __restrict__ Dvec,   // (INTER)
                     float* __restrict__ yfin) {       // (S, INTER)
  const int lane = threadIdx.x & 31;
  const int d    = lane & 15;
  const int wave = threadIdx.x >> 5;
  const int ch   = blockIdx.x * 16 + wave * 2 + (lane >> 4);

  const float Ad = -__expf(A_log[ch * DST + d]);
  const float Dh = Dvec[ch];
  float state = 0.0f;

  for (int s = 0; s < S_LEN; ++s) {
    const float dtv = dt[(size_t)s * INTER + ch];          // broadcast in L0
    const float u   = hbuf[(size_t)s * INTER + ch];
    const float Bv  = ssmp[(size_t)s * NPAD + DTR + d];
    const float Cv  = ssmp[(size_t)s * NPAD + DTR + DST + d];

    const float da = __expf(dtv * Ad);
    state = da * state + dtv * Bv * u;

    float part = state * Cv;
    part += __shfl_xor(part, 1, 32);
    part += __shfl_xor(part, 2, 32);
    part += __shfl_xor(part, 4, 32);
    part += __shfl_xor(part, 8, 32);     // full sum within each 16-lane half

    if (d == 0) {
      const float g  = proj[(size_t)s * (2 * INTER) + INTER + ch];
      const float gs = g / (1.0f + __expf(-g));
      yfin[(size_t)s * INTER + ch] = (part + u * Dh) * gs;
    }
  }
}

// ---------------------------------------------------------------------------
extern "C" void kernel_launch(void* const* d_in, const int* in_sizes, int n_in,
                              void* d_out, int out_size, void* d_ws, size_t ws_size,
                              hipStream_t stream) {
  (void)in_sizes; (void)n_in; (void)out_size; (void)ws_size;

  const float* x      = (const float*)d_in[0];   // (1, S, H)
  const float* W_in   = (const float*)d_in[1];   // (H, 2*INTER)
  const float* conv_w = (const float*)d_in[2];   // (K, 1, INTER)
  const float* conv_b = (const float*)d_in[3];   // (INTER)
  const float* W_x    = (const float*)d_in[4];   // (INTER, 96)
  const float* W_dt   = (const float*)d_in[5];   // (DTR, INTER)
  const float* b_dt   = (const float*)d_in[6];   // (INTER)
  const float* A_log  = (const float*)d_in[7];   // (INTER, DST)
  const float* Dvec   = (const float*)d_in[8];   // (INTER)
  const float* W_out  = (const float*)d_in[9];   // (INTER, H)
  float* out = (float*)d_out;                    // (S, H)

  // workspace carve-up (floats)
  float* ws   = (float*)d_ws;
  float* proj = ws;                                   // S * 2*INTER  = 8M
  float* hbuf = proj + (size_t)S_LEN * 2 * INTER;     // S * INTER    = 4M
  float* wxp  = hbuf + (size_t)S_LEN * INTER;         // INTER * 128  = 256K
  float* ssmp = wxp  + (size_t)INTER * NPAD;          // S * 128      = 256K
  float* dt   = ssmp + (size_t)S_LEN * NPAD;          // S * INTER    = 4M
  float* yfin = dt   + (size_t)S_LEN * INTER;         // S * INTER    = 4M

  const dim3 blk(256);

  // 1) proj = x @ W_in        (2048 x 1024 x 4096)
  gemm_wmma_f32<<<dim3((2 * INTER) / 64, S_LEN / 128), blk, 0, stream>>>(
      x, HID, W_in, 2 * INTER, proj, 2 * INTER, HID);

  // 2) h = silu(depthwise_conv(raw) + conv_b)
  conv_silu_kernel<<<(S_LEN * INTER) / 256, blk, 0, stream>>>(proj, conv_w, conv_b, hbuf);

  // 3) pad W_x to 128 cols
  pad_wx_kernel<<<(INTER * NPAD) / 256, blk, 0, stream>>>(W_x, wxp);

  // 4) ssm_p = h @ W_x_pad    (2048 x 2048 x 128)
  gemm_wmma_f32<<<dim3(NPAD / 64, S_LEN / 128), blk, 0, stream>>>(
      hbuf, INTER, wxp, NPAD, ssmp, NPAD, INTER);

  // 5) dt_pre = ts @ W_dt     (2048 x 64 x 2048); ts = ssm_p[:, :64], lda=128
  gemm_wmma_f32<<<dim3(INTER / 64, S_LEN / 128), blk, 0, stream>>>(
      ssmp, NPAD, W_dt, INTER, dt, INTER, DTR);

  // 6) dt = softplus(dt_pre + b_dt)
  softplus_bias_kernel<<<(S_LEN * INTER) / 256, blk, 0, stream>>>(dt, b_dt);

  // 7) fused selective scan + D-skip + gate  -> yfin
  ssm_scan_kernel<<<INTER / 16, blk, 0, stream>>>(ssmp, dt, hbuf, proj, A_log, Dvec, yfin);

  // 8) out = yfin @ W_out     (2048 x 2048 x 1024)
  gemm_wmma_f32<<<dim3(HID / 64, S_LEN / 128), blk, 0, stream>>>(
      yfin, INTER, W_out, HID, out, HID, INTER);
}